// APSDG_57011395887436
// MI455X (gfx1250) — compile-verified
//
#include <hip/hip_runtime.h>
#include <hip/hip_bf16.h>
#include <math.h>

#define D 128
#define TILE_M 16
#define LDS_STRIDE 132   // 16 rows x 132 floats: conflict-free float2 reads, 16B aligned rows

typedef float v2f __attribute__((ext_vector_type(2)));
typedef float v8f __attribute__((ext_vector_type(8)));

__device__ __forceinline__ float warp_reduce_sum(float v) {
#pragma unroll
    for (int off = 16; off > 0; off >>= 1)
        v += __shfl_xor(v, off, 32);
    return v;
}

__device__ __forceinline__ void atomic_add_f32(float* p, float v) {
    __hip_atomic_fetch_add(p, v, __ATOMIC_RELAXED, __HIP_MEMORY_SCOPE_AGENT);
}

// ---------------------------------------------------------------------------
// deg[dst[e]] += 1
// ---------------------------------------------------------------------------
__global__ void __launch_bounds__(256)
deg_kernel(const int* __restrict__ dst, float* __restrict__ deg, int E) {
    int e = blockIdx.x * blockDim.x + threadIdx.x;
    if (e < E) atomic_add_f32(&deg[dst[e]], 1.0f);
}

// ---------------------------------------------------------------------------
// Per-row radial transforms. mode 1: Poincare log-map at origin
//   scale = (2/sqrt(c)) * atanh(sqrt(c)*n) / n,  n = max(||x||, 1e-10)
// mode 2: l2norm  scale = 1/max(||x||, 1e-12)
// One wave (32 lanes) per row; 4 floats (one float4) per lane. in==out OK.
// ---------------------------------------------------------------------------
__global__ void __launch_bounds__(256)
rowscale_kernel(const float* __restrict__ in, float* __restrict__ out,
                const float* __restrict__ cptr, int mode, int n_nodes) {
    int lane = threadIdx.x & 31;
    int row  = blockIdx.x * (blockDim.x >> 5) + (threadIdx.x >> 5);
    if (row >= n_nodes) return;
    const float* rp = in + (size_t)row * D + lane * 4;
    float4 v = *(const float4*)rp;
    float ss = v.x * v.x + v.y * v.y + v.z * v.z + v.w * v.w;
    ss = warp_reduce_sum(ss);
    float nrm = sqrtf(ss);
    float scale;
    if (mode == 1) {
        float n  = fmaxf(nrm, 1e-10f);
        float sc = sqrtf(cptr[0]);
        scale = (2.0f / sc) * atanhf(sc * n) / n;
    } else {
        scale = 1.0f / fmaxf(nrm, 1e-12f);
    }
    float4 o = make_float4(v.x * scale, v.y * scale, v.z * scale, v.w * scale);
    *(float4*)(out + (size_t)row * D + lane * 4) = o;
}

// ---------------------------------------------------------------------------
// H[n, j] = sum_k A[n,k] * W[j,k] + bias[j]   (h = A @ W^T + b)
// Block: 256 threads = 8 waves. Block computes 16 rows x 128 cols.
// Wave w computes the 16x16 tile with column base w*16 using
// v_wmma_f32_16x16x4_f32, K-loop of 32 steps over D=128.
// A tile staged in LDS (stride 132 => 4r+c bank pattern, conflict free).
// ---------------------------------------------------------------------------
__global__ void __launch_bounds__(256)
gemm_wmma_kernel(const float* __restrict__ A, const float* __restrict__ W,
                 const float* __restrict__ bias, float* __restrict__ H,
                 int n_nodes) {
    __shared__ float ldsA[TILE_M * LDS_STRIDE];
    const int nodeBase = blockIdx.x * TILE_M;
    const int tid = threadIdx.x;

    // Cooperative load of 16x128 A tile: 512 float4 / 256 threads = 2 each.
#pragma unroll
    for (int p = 0; p < 2; ++p) {
        int idx  = tid + p * 256;   // float4 index 0..511
        int row  = idx >> 5;        // 32 float4 per row
        int col4 = idx & 31;
        int gr = nodeBase + row;
        float4 v = make_float4(0.f, 0.f, 0.f, 0.f);
        if (gr < n_nodes) v = *(const float4*)(A + (size_t)gr * D + col4 * 4);
        *(float4*)(&ldsA[row * LDS_STRIDE + col4 * 4]) = v;
    }
    __syncthreads();

    const int lane = tid & 31;
    const int wave = tid >> 5;
    const int lh = lane >> 4;   // K-half select for A/B fragments
    const int ln = lane & 15;   // M (for A) / N (for B, C, D)
    const int j  = wave * 16 + ln;

    const float* __restrict__ wrow = W + (size_t)j * D;   // B[k][n] = W[j][k]
    const float* __restrict__ arow = &ldsA[ln * LDS_STRIDE];

    v8f c = {0.f, 0.f, 0.f, 0.f, 0.f, 0.f, 0.f, 0.f};
#pragma unroll
    for (int k = 0; k < D; k += 4) {
        int ko = k + 2 * lh;
        v2f a; a.x = arow[ko]; a.y = arow[ko + 1];
        v2f b; b.x = wrow[ko]; b.y = wrow[ko + 1];
        // (neg_a, A, neg_b, B, c_mod, C, reuse_a, reuse_b)
        c = __builtin_amdgcn_wmma_f32_16x16x4_f32(
                false, a, false, b, (short)0, c, false, false);
    }

    const float bv = bias[j];
#pragma unroll
    for (int r = 0; r < 8; ++r) {
        int m = nodeBase + r + 8 * lh;   // C/D layout: M = r + 8*(lane/16)
        if (m < n_nodes) H[(size_t)m * D + j] = c[r] + bv;
    }
}

// ---------------------------------------------------------------------------
// acc[dst[e], :] += h[src[e], :]   one wave per edge, float4 per lane.
// h and acc (~25.6 MB each) are L2-resident, so gathers + atomics stay on-chip.
// ---------------------------------------------------------------------------
__global__ void __launch_bounds__(256)
scatter_kernel(const float* __restrict__ h, const int* __restrict__ src,
               const int* __restrict__ dst, float* __restrict__ acc, int E) {
    int lane = threadIdx.x & 31;
    int e = blockIdx.x * (blockDim.x >> 5) + (threadIdx.x >> 5);
    if (e >= E) return;
    int s = src[e], d = dst[e];
    float4 v = *(const float4*)(h + (size_t)s * D + lane * 4);
    float* ap = acc + (size_t)d * D + lane * 4;
    atomic_add_f32(ap + 0, v.x);
    atomic_add_f32(ap + 1, v.y);
    atomic_add_f32(ap + 2, v.z);
    atomic_add_f32(ap + 3, v.w);
}

// ---------------------------------------------------------------------------
// In-place: v = acc/max(deg,1); then per-channel post:
// mode 0: LeakyReLU(0.2)
// mode 1: exp-map at origin: v * tanh(sqrt(c)*n/2)/(sqrt(c)*n), n=max(||v||,1e-10)
// mode 2: l2norm
// ---------------------------------------------------------------------------
__global__ void __launch_bounds__(256)
post_kernel(float* __restrict__ acc, const float* __restrict__ deg,
            const float* __restrict__ cptr, int mode, int n_nodes) {
    int lane = threadIdx.x & 31;
    int row  = blockIdx.x * (blockDim.x >> 5) + (threadIdx.x >> 5);
    if (row >= n_nodes) return;
    float invdeg = 1.0f / fmaxf(deg[row], 1.0f);
    float* rp = acc + (size_t)row * D + lane * 4;
    float4 v = *(float4*)rp;
    v.x *= invdeg; v.y *= invdeg; v.z *= invdeg; v.w *= invdeg;
    if (mode == 0) {
        v.x = v.x > 0.f ? v.x : 0.2f * v.x;
        v.y = v.y > 0.f ? v.y : 0.2f * v.y;
        v.z = v.z > 0.f ? v.z : 0.2f * v.z;
        v.w = v.w > 0.f ? v.w : 0.2f * v.w;
    } else {
        float ss = v.x * v.x + v.y * v.y + v.z * v.z + v.w * v.w;
        ss = warp_reduce_sum(ss);
        float nrm = sqrtf(ss);
        float scale;
        if (mode == 1) {
            float n  = fmaxf(nrm, 1e-10f);
            float sc = sqrtf(cptr[0]);
            scale = tanhf(sc * n * 0.5f) / (sc * n);
        } else {
            scale = 1.0f / fmaxf(nrm, 1e-12f);
        }
        v.x *= scale; v.y *= scale; v.z *= scale; v.w *= scale;
    }
    *(float4*)rp = v;
}

// ---------------------------------------------------------------------------
extern "C" void kernel_launch(void* const* d_in, const int* in_sizes, int n_in,
                              void* d_out, int out_size, void* d_ws, size_t ws_size,
                              hipStream_t stream) {
    (void)n_in; (void)out_size; (void)ws_size;
    const int*   src   = (const int*)d_in[0];
    const int*   dst   = (const int*)d_in[1];
    const float* e_emb = (const float*)d_in[2];
    const float* b_emb = (const float*)d_in[3];
    const float* s_emb = (const float*)d_in[4];
    const float* e_W   = (const float*)d_in[5];
    const float* e_b   = (const float*)d_in[6];
    const float* b_W   = (const float*)d_in[7];
    const float* b_b   = (const float*)d_in[8];
    const float* s_W   = (const float*)d_in[9];
    const float* s_b   = (const float*)d_in[10];
    const float* b_c   = (const float*)d_in[11];
    // s_curvature (d_in[12]) is unused by the reference spherical path.

    const int E = in_sizes[0];
    const int N = in_sizes[2] / D;
    const size_t ND = (size_t)N * D;

    float* ws    = (float*)d_ws;
    float* cur_e = ws;             // layer-0 outputs
    float* cur_b = ws + ND;
    float* cur_s = ws + 2 * ND;
    float* tmp   = ws + 3 * ND;    // pre-transformed GEMM input
    float* h     = ws + 4 * ND;    // GEMM output / scatter source
    float* deg   = ws + 5 * ND;    // N floats

    float* out_e = (float*)d_out;
    float* out_b = out_e + ND;
    float* out_s = out_e + 2 * ND;

    const dim3 blk(256);
    const int gemmGrid  = (N + TILE_M - 1) / TILE_M;
    const int rowGrid   = (N + 7) / 8;
    const int edgeWGrid = (E + 7) / 8;
    const int edgeGrid  = (E + 255) / 256;

    // Degree histogram (shared by all channels/layers)
    hipMemsetAsync(deg, 0, (size_t)N * sizeof(float), stream);
    deg_kernel<<<edgeGrid, blk, 0, stream>>>(dst, deg, E);

    for (int l = 0; l < 2; ++l) {
        const float* in_e = (l == 0) ? e_emb : cur_e;
        const float* in_b = (l == 0) ? b_emb : cur_b;
        const float* in_s = (l == 0) ? s_emb : cur_s;
        float* a_e = (l == 0) ? cur_e : out_e;
        float* a_b = (l == 0) ? cur_b : out_b;
        float* a_s = (l == 0) ? cur_s : out_s;
        const size_t wo = (size_t)l * D * D;
        const size_t bo = (size_t)l * D;

        // ---- Euclidean: GEMM -> mean agg -> LeakyReLU ----
        gemm_wmma_kernel<<<gemmGrid, blk, 0, stream>>>(in_e, e_W + wo, e_b + bo, h, N);
        hipMemsetAsync(a_e, 0, ND * sizeof(float), stream);
        scatter_kernel<<<edgeWGrid, blk, 0, stream>>>(h, src, dst, a_e, E);
        post_kernel<<<rowGrid, blk, 0, stream>>>(a_e, deg, nullptr, 0, N);

        // ---- Hyperbolic: log-map -> GEMM -> mean agg -> exp-map ----
        rowscale_kernel<<<rowGrid, blk, 0, stream>>>(in_b, tmp, b_c, 1, N);
        gemm_wmma_kernel<<<gemmGrid, blk, 0, stream>>>(tmp, b_W + wo, b_b + bo, h, N);
        hipMemsetAsync(a_b, 0, ND * sizeof(float), stream);
        scatter_kernel<<<edgeWGrid, blk, 0, stream>>>(h, src, dst, a_b, E);
        post_kernel<<<rowGrid, blk, 0, stream>>>(a_b, deg, b_c, 1, N);

        // ---- Spherical: l2norm -> GEMM -> l2norm -> mean agg -> l2norm ----
        rowscale_kernel<<<rowGrid, blk, 0, stream>>>(in_s, tmp, nullptr, 2, N);
        gemm_wmma_kernel<<<gemmGrid, blk, 0, stream>>>(tmp, s_W + wo, s_b + bo, h, N);
        rowscale_kernel<<<rowGrid, blk, 0, stream>>>(h, h, nullptr, 2, N);
        hipMemsetAsync(a_s, 0, ND * sizeof(float), stream);
        scatter_kernel<<<edgeWGrid, blk, 0, stream>>>(h, src, dst, a_s, E);
        post_kernel<<<rowGrid, blk, 0, stream>>>(a_s, deg, nullptr, 2, N);
    }
}